// LSTM_49675591745926
// MI455X (gfx1250) — compile-verified
//
#include <hip/hip_runtime.h>
#include <math.h>

#define V_  50257
#define H_  768
#define B_  16
#define T_  128
#define H2_ 1536
#define H4_ 3072
#define BT_ 2048
#define NVT ((V_ + 15) / 16)      /* 3142 vocab tiles */
#define VTPW 4                    /* vocab tiles per wave in K3 */
#define NTPW 4                    /* n tiles per wave in K1 */

/* persistent LSTM kernel config: 96 waves x 2 tiles = 192 gate tiles */
#define NB2    12
#define THR2   256
#define WAVES2 (NB2 * THR2 / 32)  /* 96 waves */
#define GTILES (H4_ / 16)         /* 192 gate tiles per step */

typedef __attribute__((ext_vector_type(2))) float v2f;
typedef __attribute__((ext_vector_type(8))) float v8f;

__device__ __forceinline__ float sigmf(float x) { return 1.0f / (1.0f + expf(-x)); }

/* ------------------------------------------------------------------ */
/* device-wide barrier (L2 atomics); bar[0]=arrive count, bar[1]=gen  */
__device__ __forceinline__ void grid_barrier(int* bar, int nblk) {
  __syncthreads();
  if (threadIdx.x == 0) {
    int gen = __hip_atomic_load(&bar[1], __ATOMIC_ACQUIRE, __HIP_MEMORY_SCOPE_AGENT);
    int arr = __hip_atomic_fetch_add(&bar[0], 1, __ATOMIC_ACQ_REL, __HIP_MEMORY_SCOPE_AGENT);
    if (arr == nblk - 1) {
      __hip_atomic_store(&bar[0], 0, __ATOMIC_RELAXED, __HIP_MEMORY_SCOPE_AGENT);
      __hip_atomic_fetch_add(&bar[1], 1, __ATOMIC_RELEASE, __HIP_MEMORY_SCOPE_AGENT);
    } else {
      while (__hip_atomic_load(&bar[1], __ATOMIC_ACQUIRE, __HIP_MEMORY_SCOPE_AGENT) == gen)
        __builtin_amdgcn_s_sleep(1);
    }
  }
  __syncthreads();
}

/* ------------------------------------------------------------------ */
/* K0: init h/c state, last-attended index, barrier counters          */
__global__ __launch_bounds__(256)
void k_init(const int* __restrict__ mask, const float* __restrict__ h0,
            const float* __restrict__ c0, float* __restrict__ wsh,
            float* __restrict__ wsc, int* __restrict__ last, int* __restrict__ bar) {
  int tid = blockIdx.x * blockDim.x + threadIdx.x;
  for (int i = tid; i < B_ * H_; i += gridDim.x * blockDim.x) {
    wsh[i] = h0[i];
    wsc[i] = c0[i];
  }
  if (blockIdx.x == 0 && threadIdx.x < B_) {
    int s = 0;
    for (int t = 0; t < T_; ++t) s += mask[threadIdx.x * T_ + t];
    last[threadIdx.x] = s - 1;
  }
  if (blockIdx.x == 0 && threadIdx.x == 0) { bar[0] = 0; bar[1] = 0; }
}

/* ------------------------------------------------------------------ */
/* K1: xg[bt,n] = [emb(ids[bt]) | append(b)] . W_ih[n,:] + b_ih[n]    */
/* one 16x(16*NTPW) block per wave: A fragment reused NTPW times      */
__global__ __launch_bounds__(256)
void k_xg(const int* __restrict__ ids, const float* __restrict__ emb,
          const float* __restrict__ app, const float* __restrict__ Wih,
          const float* __restrict__ bih, float* __restrict__ xg) {
  const int lane = threadIdx.x & 31;
  const int wid  = (blockIdx.x * blockDim.x + threadIdx.x) >> 5;
  const int NGRP = GTILES / NTPW;            /* 48 n-groups */
  const int btT = wid / NGRP, ng = wid % NGRP;
  const int bt0 = btT * 16, nT0 = ng * NTPW;
  const int r  = lane & 15;   /* A row (bt), also B/C column (n) */
  const int kh = lane >> 4;   /* which K half this lane holds    */
  const int b  = bt0 / T_;    /* 16 | T so whole tile shares b   */
  const int rid = ids[bt0 + r];
  const float* wrow[NTPW];
#pragma unroll
  for (int q = 0; q < NTPW; ++q)
    wrow[q] = Wih + (size_t)((nT0 + q) * 16 + r) * H2_;
  v8f acc[NTPW] = {};
  for (int k0 = 0; k0 < H2_; k0 += 4) {
    int kk = k0 + 2 * kh;
    v2f a;
    if (kk < H_) {
      a.x = emb[(size_t)rid * H_ + kk];
      a.y = emb[(size_t)rid * H_ + kk + 1];
    } else {
      a.x = app[b * H_ + kk - H_];
      a.y = app[b * H_ + kk - H_ + 1];
    }
#pragma unroll
    for (int q = 0; q < NTPW; ++q) {
      v2f bb;
      bb.x = wrow[q][kk];
      bb.y = wrow[q][kk + 1];
      acc[q] = __builtin_amdgcn_wmma_f32_16x16x4_f32(false, a, false, bb,
                                                     (short)0, acc[q], false, false);
    }
  }
#pragma unroll
  for (int q = 0; q < NTPW; ++q) {
    int nn = (nT0 + q) * 16 + r;
    float bias = bih[nn];
#pragma unroll
    for (int j = 0; j < 8; ++j) {
      int m = j + 8 * kh;
      xg[(size_t)(bt0 + m) * H4_ + nn] = acc[q][j] + bias;
    }
  }
}

/* ------------------------------------------------------------------ */
/* K2: persistent LSTM scan. Per step: gates = h@W_hhT (+xg_t +b_hh)  */
/* via WMMA (M=16 batch rows); each wave owns 2 tiles sharing A-load. */
__global__ __launch_bounds__(THR2)
void k_lstm(const float* __restrict__ xg, const float* __restrict__ Whh,
            const float* __restrict__ bhh, float* __restrict__ wsh,
            float* __restrict__ wsc, float* __restrict__ gates,
            const int* __restrict__ last, int* __restrict__ bar,
            float* __restrict__ out_h, float* __restrict__ out_selc) {
  const int lane = threadIdx.x & 31;
  const int wid  = (blockIdx.x * THR2 + threadIdx.x) >> 5;
  const int gtid = blockIdx.x * THR2 + threadIdx.x;
  const int r  = lane & 15;
  const int kh = lane >> 4;
  const int nn0 = (wid * 2) * 16 + r;        /* tile pair: 2*wid, 2*wid+1 */
  const int nn1 = nn0 + 16;
  const float* wrow0 = Whh + (size_t)nn0 * H_;
  const float* wrow1 = Whh + (size_t)nn1 * H_;
  const float bias0 = bhh[nn0];
  const float bias1 = bhh[nn1];
  for (int t = 0; t < T_; ++t) {
    v8f acc0 = {}, acc1 = {};
    for (int k0 = 0; k0 < H_; k0 += 4) {
      int kk = k0 + 2 * kh;
      v2f a, b0, b1;
      a.x = wsh[r * H_ + kk];
      a.y = wsh[r * H_ + kk + 1];
      b0.x = wrow0[kk];
      b0.y = wrow0[kk + 1];
      b1.x = wrow1[kk];
      b1.y = wrow1[kk + 1];
      acc0 = __builtin_amdgcn_wmma_f32_16x16x4_f32(false, a, false, b0,
                                                   (short)0, acc0, false, false);
      acc1 = __builtin_amdgcn_wmma_f32_16x16x4_f32(false, a, false, b1,
                                                   (short)0, acc1, false, false);
    }
#pragma unroll
    for (int j = 0; j < 8; ++j) {
      int m = j + 8 * kh; /* batch row */
      const float* xr = xg + (size_t)(m * T_ + t) * H4_;
      gates[m * H4_ + nn0] = acc0[j] + xr[nn0] + bias0;
      gates[m * H4_ + nn1] = acc1[j] + xr[nn1] + bias1;
    }
    grid_barrier(bar, NB2);
    /* elementwise LSTM cell update */
    for (int e = gtid; e < B_ * H_; e += NB2 * THR2) {
      int b = e / H_, hh = e - b * H_;
      const float* g = gates + b * H4_;
      float ig = sigmf(g[hh]);
      float fg = sigmf(g[H_ + hh]);
      float gg = tanhf(g[2 * H_ + hh]);
      float og = sigmf(g[3 * H_ + hh]);
      float c = fg * wsc[e] + ig * gg;
      float h = og * tanhf(c);
      wsc[e] = c;
      wsh[e] = h;
      out_h[(size_t)(b * T_ + t) * H_ + hh] = h;
      if (t == last[b]) out_selc[e] = c;
    }
    grid_barrier(bar, NB2);
  }
}

/* ------------------------------------------------------------------ */
/* K3: logits = hiddens @ lm_wT + lm_b  (dominant GEMM, 158 GFLOP).   */
/* Each wave: 1 bt-tile x VTPW vocab tiles; A fragment reused VTPW x. */
/* blockIdx.x = bt-tile (fast axis) -> concurrent blocks share a      */
/* ~1.5MB lm_w slice in L2 (lm_w fits the 192MB L2 entirely).         */
__global__ __launch_bounds__(256)
void k_logits(const float* __restrict__ hid, const float* __restrict__ lmw,
              const float* __restrict__ lmb, float* __restrict__ logits) {
  const int lane = threadIdx.x & 31;
  const int wIn  = threadIdx.x >> 5;
  const int vT0  = (blockIdx.y * 8 + wIn) * VTPW;
  if (vT0 >= NVT) return;                 /* wave-uniform */
  const int bt0 = blockIdx.x * 16;
  const int r  = lane & 15;
  const int kh = lane >> 4;
  const float* arow = hid + (size_t)(bt0 + r) * H_;
  const float* brow[VTPW];
#pragma unroll
  for (int q = 0; q < VTPW; ++q) {
    int v  = (vT0 + q) * 16 + r;
    int vc = v < V_ ? v : V_ - 1;         /* clamp pad columns */
    brow[q] = lmw + (size_t)vc * H_;
  }
  v8f acc[VTPW] = {};
  for (int k0 = 0; k0 < H_; k0 += 4) {
    int kk = k0 + 2 * kh;
    v2f a;
    a.x = arow[kk];
    a.y = arow[kk + 1];
#pragma unroll
    for (int q = 0; q < VTPW; ++q) {
      v2f bb;
      bb.x = brow[q][kk];
      bb.y = brow[q][kk + 1];
      acc[q] = __builtin_amdgcn_wmma_f32_16x16x4_f32(false, a, false, bb,
                                                     (short)0, acc[q], false, false);
    }
  }
#pragma unroll
  for (int q = 0; q < VTPW; ++q) {
    int v = (vT0 + q) * 16 + r;
    if (v < V_) {
      float bias = lmb[v];
#pragma unroll
      for (int j = 0; j < 8; ++j) {
        int m = j + 8 * kh;
        logits[(size_t)(bt0 + m) * V_ + v] = acc[q][j] + bias;
      }
    }
  }
}

/* ------------------------------------------------------------------ */
/* K4: fused log_softmax per row (max, sum-exp, subtract); the 200KB  */
/* row stays L2-resident across the three passes.                     */
__global__ __launch_bounds__(256)
void k_lsm(float* __restrict__ logits) {
  __shared__ float red[256];
  float* p = logits + (size_t)blockIdx.x * V_;
  const int tid = threadIdx.x;
  float m = -INFINITY;
  for (int i = tid; i < V_; i += 256) m = fmaxf(m, p[i]);
  red[tid] = m;
  __syncthreads();
  for (int s = 128; s > 0; s >>= 1) {
    if (tid < s) red[tid] = fmaxf(red[tid], red[tid + s]);
    __syncthreads();
  }
  float rmax = red[0];
  __syncthreads();
  float sum = 0.f;
  for (int i = tid; i < V_; i += 256) sum += expf(p[i] - rmax);
  red[tid] = sum;
  __syncthreads();
  for (int s = 128; s > 0; s >>= 1) {
    if (tid < s) red[tid] += red[tid + s];
    __syncthreads();
  }
  float lse = rmax + logf(red[0]);
  for (int i = tid; i < V_; i += 256) p[i] = p[i] - lse;
}

/* ------------------------------------------------------------------ */
extern "C" void kernel_launch(void* const* d_in, const int* in_sizes, int n_in,
                              void* d_out, int out_size, void* d_ws, size_t ws_size,
                              hipStream_t stream) {
  const int*   ids  = (const int*)d_in[0];
  const int*   mask = (const int*)d_in[1];
  const float* h0   = (const float*)d_in[2];
  const float* c0   = (const float*)d_in[3];
  const float* app  = (const float*)d_in[4];
  const float* emb  = (const float*)d_in[5];
  const float* Wih  = (const float*)d_in[6];
  const float* Whh  = (const float*)d_in[7];
  const float* bih  = (const float*)d_in[8];
  const float* bhh  = (const float*)d_in[9];
  const float* lmw  = (const float*)d_in[10];
  const float* lmb  = (const float*)d_in[11];

  float* out = (float*)d_out;
  float* out_h    = out;                          /* [B,T,H]  */
  float* out_selc = out + (size_t)B_ * T_ * H_;   /* [B,H]    */
  float* out_lp   = out_selc + (size_t)B_ * H_;   /* [B,T,V]  */

  /* workspace carve (~25.5 MB) */
  char* w = (char*)d_ws;
  float* ws_xg = (float*)w;  w += (size_t)BT_ * H4_ * sizeof(float);
  float* ws_h  = (float*)w;  w += (size_t)B_ * H_ * sizeof(float);
  float* ws_c  = (float*)w;  w += (size_t)B_ * H_ * sizeof(float);
  float* ws_g  = (float*)w;  w += (size_t)B_ * H4_ * sizeof(float);
  int*   ws_last = (int*)w;  w += 256;
  int*   ws_bar  = (int*)w;

  k_init<<<dim3(48), 256, 0, stream>>>(mask, h0, c0, ws_h, ws_c, ws_last, ws_bar);
  /* K1: 128 bt-tiles x 48 n-groups = 6144 waves = 768 blocks */
  k_xg<<<dim3((BT_ / 16) * (GTILES / NTPW) / 8), 256, 0, stream>>>(ids, emb, app,
                                                                   Wih, bih, ws_xg);
  k_lstm<<<dim3(NB2), THR2, 0, stream>>>(ws_xg, Whh, bhh, ws_h, ws_c, ws_g,
                                         ws_last, ws_bar, out_h, out_selc);
  /* K3: y covers ceil(3142 / (8 waves * 4 tiles)) = 99 groups */
  k_logits<<<dim3(128, (NVT + 8 * VTPW - 1) / (8 * VTPW)), 256, 0, stream>>>(
      out_h, lmw, lmb, out_lp);
  k_lsm<<<dim3(BT_), 256, 0, stream>>>(out_lp);
}